// BaselineAttention_67190468379231
// MI455X (gfx1250) — compile-verified
//
#include <hip/hip_runtime.h>

typedef float v2f __attribute__((ext_vector_type(2)));
typedef float v8f __attribute__((ext_vector_type(8)));

// D = A(16x4 f32) * B(4x16 f32) + C(16x16 f32), full wave32 op.
__device__ __forceinline__ v8f wmma_f32(v2f a, v2f b, v8f c) {
    // (neg_a, A, neg_b, B, c_mod, C, reuse_a, reuse_b)
    return __builtin_amdgcn_wmma_f32_16x16x4_f32(false, a, false, b, (short)0, c,
                                                 false, false);
}

// ---------------------------------------------------------------------------
// fp32 WMMA GEMM:  O[b] = W (M x K, row-major) * X[b] (K x N)
// One wave computes a 16(M) x 64(N) strip: 4 adjacent 16x16 tiles sharing the
// A operand each K4 step. All loads of a K-step are issued before any WMMA of
// that step so the scheduler can clause them and drain s_wait_loadcnt
// incrementally instead of stalling to 0 per WMMA.
//
// A-operand layout (ISA 7.12.2, 32-bit A 16x4): lane l, vgpr j holds
//   A[l&15][2*(l>>4) + j]
// B-operand (4x16), mirrored: lane l, vgpr j holds B[2*(l>>4)+j][l&15]
// C/D (16x16 f32): vgpr r, lane l holds D[r + 8*(l>>4)][l&15]
// ---------------------------------------------------------------------------
__global__ __launch_bounds__(256) void gemm_f32_wmma(
    const float* __restrict__ W,   // M x K
    const float* __restrict__ X,   // batches x K x N
    float* __restrict__ O,         // batches x M x N
    int Mtiles, int K, int N, int batches)
{
    const int wid  = blockIdx.x * (blockDim.x >> 5) + (threadIdx.x >> 5);
    const int lane = threadIdx.x & 31;
    const int lm   = lane & 15;
    const int hi   = lane >> 4;

    const int Nblks = N >> 6;            // 64-wide N blocks
    int t = wid;
    const int nb = t % Nblks; t /= Nblks;
    const int mt = t % Mtiles; t /= Mtiles;
    const int b  = t;
    if (b >= batches) return;

    const int m0 = mt << 4;
    const int n0 = nb << 6;

    const float* Wp = W + (size_t)(m0 + lm) * K + 2 * hi;
    const float* Xp = X + (size_t)b * K * N + (size_t)(2 * hi) * N + n0 + lm;

    v8f acc[4] = {};
#pragma unroll 2
    for (int k0 = 0; k0 < K; k0 += 4) {
        // ---- batch all loads for this K-step ----
        v2f a = *(const v2f*)(Wp + k0);            // A[lm][k0+2hi .. +1]
        const float* xr0 = Xp + (size_t)k0 * N;    // row k0+2hi
        const float* xr1 = xr0 + N;                // row k0+2hi+1
        v2f bb[4];
#pragma unroll
        for (int j = 0; j < 4; ++j) {
            bb[j].x = xr0[16 * j];                 // B[k0+2hi  ][16j+lm]
            bb[j].y = xr1[16 * j];                 // B[k0+2hi+1][16j+lm]
        }
        // ---- then the 4 WMMAs ----
#pragma unroll
        for (int j = 0; j < 4; ++j)
            acc[j] = wmma_f32(a, bb[j], acc[j]);
    }

    float* Op = O + ((size_t)b * (Mtiles << 4) + m0) * N + n0;
#pragma unroll
    for (int j = 0; j < 4; ++j)
#pragma unroll
        for (int r = 0; r < 8; ++r)
            Op[(size_t)(r + 8 * hi) * N + 16 * j + lm] = acc[j][r];
}

// ---------------------------------------------------------------------------
// Flash attention, one wave per (b, h, 32-row n-slab) = two 16-row tiles that
// share every K-tile and V-tile operand. K and V operands for the m-slab are
// loaded up front so V latency hides under the S WMMAs + softmax VALU work.
// QKV layout in workspace: (B, 768, N); q rows [0,256), k [256,512), v [512,768).
//   S[n][m] = (1/sqrt(32)) * sum_d q[d][n] * k[d][m]
//   O[n][d] = sum_m softmax(S)[n][m] * v[d][m]
// Online softmax; P converted D-layout -> A-layout through LDS.
// ---------------------------------------------------------------------------
__global__ __launch_bounds__(32) void flash_attn_wmma(
    const float* __restrict__ QKV, float* __restrict__ AO)
{
    __shared__ float lds[2 * 256];     // two 16x16 P tiles

    const int lane = threadIdx.x & 31;
    const int lm   = lane & 15;
    const int hi   = lane >> 4;

    int t = blockIdx.x;
    const int ns = t & 63; t >>= 6;    // 64 slabs of 32 rows
    const int h  = t & 7;  t >>= 3;    // 8 heads
    const int b  = t;                  // 4 batches
    const int n0 = ns << 5;
    const size_t N = 2048;

    const float* q = QKV + ((size_t)b * 768 +       (size_t)h * 32) * N;
    const float* k = QKV + ((size_t)b * 768 + 256 + (size_t)h * 32) * N;
    const float* v = QKV + ((size_t)b * 768 + 512 + (size_t)h * 32) * N;

    // Q tiles in A-layout registers: qa[u][dc] covers rows n0+16u.., d = 4dc..4dc+3
    v2f qa[2][8];
#pragma unroll
    for (int u = 0; u < 2; ++u)
#pragma unroll
        for (int dc = 0; dc < 8; ++dc) {
            qa[u][dc].x = q[(size_t)(4 * dc + 2 * hi)     * N + n0 + 16 * u + lm];
            qa[u][dc].y = q[(size_t)(4 * dc + 2 * hi + 1) * N + n0 + 16 * u + lm];
        }

    v8f o[2][2] = {};                  // [tile u][d-half]
    float rmax[2][8], rsum[2][8];
#pragma unroll
    for (int u = 0; u < 2; ++u)
#pragma unroll
        for (int r = 0; r < 8; ++r) { rmax[u][r] = -3.4e38f; rsum[u][r] = 0.0f; }
    const float inv_scale = 0.17677669529663687f;  // 1/sqrt(32)

    for (int m0 = 0; m0 < 2048; m0 += 16) {
        // ---- K tile operands (shared by both Q tiles) ----
        v2f kb[8];
#pragma unroll
        for (int dc = 0; dc < 8; ++dc) {
            kb[dc].x = k[(size_t)(4 * dc + 2 * hi)     * N + m0 + lm];
            kb[dc].y = k[(size_t)(4 * dc + 2 * hi + 1) * N + m0 + lm];
        }
        // ---- V tile operands, issued early: latency hides under S + softmax ----
        v2f vb0[4], vb1[4];               // B[m][d] = v[d][m]
#pragma unroll
        for (int mc = 0; mc < 4; ++mc) {
            const size_t mm = (size_t)m0 + 4 * mc + 2 * hi;
            vb0[mc].x = v[(size_t)lm * N + mm];
            vb0[mc].y = v[(size_t)lm * N + mm + 1];
            vb1[mc].x = v[(size_t)(16 + lm) * N + mm];
            vb1[mc].y = v[(size_t)(16 + lm) * N + mm + 1];
        }

#pragma unroll
        for (int u = 0; u < 2; ++u) {
            v8f s = {};
#pragma unroll
            for (int dc = 0; dc < 8; ++dc)
                s = wmma_f32(qa[u][dc], kb[dc], s);

            // online softmax over this 16-wide m slab
            float cf[8];
#pragma unroll
            for (int r = 0; r < 8; ++r) {
                float x = s[r] * inv_scale;   // row n0+16u+r+8hi, col m0+lm
                float tmax = x;               // reduce over the 16-lane N group
                tmax = fmaxf(tmax, __shfl_xor(tmax, 1));
                tmax = fmaxf(tmax, __shfl_xor(tmax, 2));
                tmax = fmaxf(tmax, __shfl_xor(tmax, 4));
                tmax = fmaxf(tmax, __shfl_xor(tmax, 8));
                float nm = fmaxf(rmax[u][r], tmax);
                cf[r]      = __expf(rmax[u][r] - nm);
                rmax[u][r] = nm;
                float e = __expf(x - nm);
                float ts = e;
                ts += __shfl_xor(ts, 1);
                ts += __shfl_xor(ts, 2);
                ts += __shfl_xor(ts, 4);
                ts += __shfl_xor(ts, 8);
                rsum[u][r] = rsum[u][r] * cf[r] + ts;
                lds[u * 256 + (r + 8 * hi) * 16 + lm] = e;   // P row-major [n][m]
            }
#pragma unroll
            for (int r = 0; r < 8; ++r) { o[u][0][r] *= cf[r]; o[u][1][r] *= cf[r]; }
        }

        asm volatile("s_wait_dscnt 0" ::: "memory");  // P stores -> A-layout reads

        // ---- O += P * V^T : V operands already resident, only LDS pa reads here
#pragma unroll
        for (int mc = 0; mc < 4; ++mc) {
            v2f pa0 = *(const v2f*)&lds[      lm * 16 + 4 * mc + 2 * hi];
            v2f pa1 = *(const v2f*)&lds[256 + lm * 16 + 4 * mc + 2 * hi];
            o[0][0] = wmma_f32(pa0, vb0[mc], o[0][0]);
            o[0][1] = wmma_f32(pa0, vb1[mc], o[0][1]);
            o[1][0] = wmma_f32(pa1, vb0[mc], o[1][0]);
            o[1][1] = wmma_f32(pa1, vb1[mc], o[1][1]);
        }
    }

    // ---- normalize and store: AO layout (B, 256, N), row = h*32 + d ----
    float* ao = AO + ((size_t)b * 256 + (size_t)h * 32) * N + n0;
#pragma unroll
    for (int u = 0; u < 2; ++u)
#pragma unroll
        for (int r = 0; r < 8; ++r) {
            float inv = 1.0f / rsum[u][r];
            ao[(size_t)lm * N        + 16 * u + r + 8 * hi] = o[u][0][r] * inv;
            ao[(size_t)(16 + lm) * N + 16 * u + r + 8 * hi] = o[u][1][r] * inv;
        }
}

// ---------------------------------------------------------------------------
extern "C" void kernel_launch(void* const* d_in, const int* in_sizes, int n_in,
                              void* d_out, int out_size, void* d_ws, size_t ws_size,
                              hipStream_t stream)
{
    const float* x      = (const float*)d_in[0];  // (4, 256, 2048)
    const float* w_qkv  = (const float*)d_in[1];  // (768, 256)
    const float* w_proj = (const float*)d_in[2];  // (256, 256)
    float* out = (float*)d_out;                   // (4, 256, 2048)

    const int B = 4, C = 256, N = 2048;

    float* qkv = (float*)d_ws;                    // B*768*N floats (25.2 MB)
    float* ao  = qkv + (size_t)B * 3 * C * N;     // B*256*N floats ( 8.4 MB)

    // 1) qkv = w_qkv @ x : M=768 (48 tiles), K=256, 16x64 strip per wave
    {
        const int Mtiles = 48;
        const int waves  = B * Mtiles * (N >> 6); // 6144
        gemm_f32_wmma<<<waves / 8, 256, 0, stream>>>(w_qkv, x, qkv, Mtiles, C, N, B);
    }

    // 2) fused attention per (b, h, 32-row slab): 4*8*64 = 2048 waves
    flash_attn_wmma<<<B * 8 * (N >> 5), 32, 0, stream>>>(qkv, ao);

    // 3) out = w_proj @ ao : M=256 (16 tiles), K=256
    {
        const int Mtiles = 16;
        const int waves  = B * Mtiles * (N >> 6); // 2048
        gemm_f32_wmma<<<waves / 8, 256, 0, stream>>>(w_proj, ao, out, Mtiles, C, N, B);
    }
}